// CeptaContextBlock_53300544143617
// MI455X (gfx1250) — compile-verified
//
#include <hip/hip_runtime.h>

// ---------------------------------------------------------------------------
// CeptaContextBlock for MI455X (gfx1250, wave32)
//  - W_U/W_V folded into W_eff  => 4 large + 2 small f32 WMMA GEMMs + scan
//  - GEMMs: v_wmma_f32_16x16x4_f32, double-buffered LDS staged with
//    global_load_async_to_lds_b128 (ASYNCcnt) via inline asm
// ---------------------------------------------------------------------------

typedef float v2f __attribute__((ext_vector_type(2)));
typedef float v8f __attribute__((ext_vector_type(8)));

#define Bv   4
#define Tv   4096
#define Dv   1024
#define Pv   512
#define PRv  64
#define Mv   (Bv * Tv)          // 16384 rows
#define NCHK 64                 // scan chunks per sequence
#define CHKL (Tv / NCHK)        // 64 steps per chunk

constexpr int BM = 128;         // block tile M
constexpr int BN = 64;          // block tile N
constexpr int KT = 32;          // K staged per LDS round (8 wmma k-steps)
constexpr int LDA_S = 36;       // padded LDS stride for A (16B aligned, conflict-free)
constexpr int LDB_S = 68;       // padded LDS stride for B (16B aligned)

enum { ACT_NONE = 0, ACT_SIGMOID = 1 };

#if defined(__gfx1250__)
#define USE_ASYNC_LDS 1
#else
#define USE_ASYNC_LDS 0
#endif

// copy 16 bytes global -> LDS.
// gfx1250: GLOBAL_LOAD_ASYNC_TO_LDS_B128 (tracked by ASYNCcnt). The LDS
// destination VGPR holds the DS byte offset == low 32 bits of the generic
// __shared__ pointer (flat aperture rule, ISA 10.2).
__device__ __forceinline__ void copy16(const float* g, float* l)
{
#if USE_ASYNC_LDS
    const unsigned lds_off = (unsigned)(uintptr_t)l;
    asm volatile("global_load_async_to_lds_b128 %0, %1, off"
                 :
                 : "v"(lds_off), "v"(g)
                 : "memory");
#else
    *(float4*)l = *(const float4*)g;
#endif
}

__device__ __forceinline__ void stage_wait()
{
#if USE_ASYNC_LDS
    asm volatile("s_wait_asynccnt 0x0" ::: "memory");
#endif
}

// ---------------------------------------------------------------------------
// W_eff[d,p] = W_toP[d,p] + sum_a W_U[d,p,a] * W_V[p,a]
// ---------------------------------------------------------------------------
__global__ __launch_bounds__(256) void weff_kernel(
    const float* __restrict__ WtoP, const float* __restrict__ WU,
    const float* __restrict__ WV, float* __restrict__ We)
{
    int g = blockIdx.x * 256 + threadIdx.x;        // over D*P
    int p = g & (Pv - 1);
    float4 wu = ((const float4*)WU)[g];            // (d*P+p)*4 contiguous
    float4 wv = ((const float4*)WV)[p];            // p*4 contiguous
    We[g] = WtoP[g] + wu.x * wv.x + wu.y * wv.y + wu.z * wv.z + wu.w * wv.w;
}

// ---------------------------------------------------------------------------
// Generic f32 WMMA GEMM: Out[M,N] = act(A[M,K] @ Bw[K,N] (+ Ini[M,N]))
// 256 threads = 8 waves laid out 4(M) x 2(N); each wave: 2x2 WMMA tiles
// (32x32). K staged 32 at a time into double-buffered LDS via async copies.
// Uniform control flow => EXEC all ones for WMMA.
// ---------------------------------------------------------------------------
template <int ACT, bool INIT>
__global__ __launch_bounds__(256) void gemm_wmma(
    const float* __restrict__ A, const float* __restrict__ Bw,
    const float* __restrict__ Ini, float* __restrict__ Out,
    int M, int N, int K)
{
    __shared__ float As[2 * BM * LDA_S];
    __shared__ float Bs[2 * KT * LDB_S];

    const int tid  = threadIdx.x;
    const int lane = tid & 31;
    const int wave = tid >> 5;
    const int wm   = wave & 3;       // 4 waves along M
    const int wn   = wave >> 2;      // 2 waves along N
    const int half = lane >> 4;      // 0: lanes 0-15, 1: lanes 16-31
    const int lm   = lane & 15;
    const int m0   = blockIdx.y * BM;
    const int n0   = blockIdx.x * BN;

    // stage one K-tile (A: 128x32, B: 32x64) into LDS buffer `buf`
    auto stage = [&](int k0, int buf) {
        float* Ad = As + buf * (BM * LDA_S);
        float* Bd = Bs + buf * (KT * LDB_S);
#pragma unroll
        for (int it = 0; it < 4; ++it) {           // 1024 float4 over 256 thr
            const int f = tid + 256 * it;
            const int row = f >> 3, col = (f & 7) << 2;
            copy16(A + (m0 + row) * K + k0 + col, Ad + row * LDA_S + col);
        }
#pragma unroll
        for (int it = 0; it < 2; ++it) {           // 512 float4 over 256 thr
            const int f = tid + 256 * it;
            const int row = f >> 4, col = (f & 15) << 2;
            copy16(Bw + (k0 + row) * N + n0 + col, Bd + row * LDB_S + col);
        }
    };

    v8f c[2][2];
#pragma unroll
    for (int i = 0; i < 2; ++i)
#pragma unroll
        for (int j = 0; j < 2; ++j) {
            if (INIT) {
                const int mb = m0 + wm * 32 + i * 16 + 8 * half;
                const int nb = n0 + wn * 32 + j * 16 + lm;
#pragma unroll
                for (int v = 0; v < 8; ++v)
                    c[i][j][v] = Ini[(mb + v) * N + nb];
            } else {
#pragma unroll
                for (int v = 0; v < 8; ++v) c[i][j][v] = 0.0f;
            }
        }

    const int ntiles = K / KT;
    stage(0, 0);

    for (int it = 0; it < ntiles; ++it) {
        const int cur = it & 1;
        // my copies into buf `cur` are done; barrier also proves every wave
        // finished reading buf `cur^1` last round before we overwrite it.
        stage_wait();
        __syncthreads();
        if (it + 1 < ntiles) stage((it + 1) * KT, cur ^ 1);

        const float* Asb = As + cur * (BM * LDA_S);
        const float* Bsb = Bs + cur * (KT * LDB_S);
#pragma unroll
        for (int kk = 0; kk < KT / 4; ++kk) {
            // A 16x4 layout: lanes0-15 hold K=0,1; lanes16-31 hold K=2,3
            const int kb = kk * 4 + 2 * half;
            v2f a[2], b[2];
#pragma unroll
            for (int i = 0; i < 2; ++i) {
                const int m = wm * 32 + i * 16 + lm;
                a[i].x = Asb[m * LDA_S + kb];
                a[i].y = Asb[m * LDA_S + kb + 1];
            }
#pragma unroll
            for (int j = 0; j < 2; ++j) {
                const int n = wn * 32 + j * 16 + lm;
                b[j].x = Bsb[kb * LDB_S + n];
                b[j].y = Bsb[(kb + 1) * LDB_S + n];
            }
#pragma unroll
            for (int i = 0; i < 2; ++i)
#pragma unroll
                for (int j = 0; j < 2; ++j)
                    c[i][j] = __builtin_amdgcn_wmma_f32_16x16x4_f32(
                        false, a[i], false, b[j], (short)0, c[i][j], false, false);
        }
        __syncthreads();
    }

    // --- epilogue: C/D layout VGPR v -> row (v + 8*half), col lm ---
#pragma unroll
    for (int i = 0; i < 2; ++i)
#pragma unroll
        for (int j = 0; j < 2; ++j) {
            const int mb = m0 + wm * 32 + i * 16 + 8 * half;
            const int nb = n0 + wn * 32 + j * 16 + lm;
#pragma unroll
            for (int v = 0; v < 8; ++v) {
                float val = c[i][j][v];
                if (ACT == ACT_SIGMOID) val = 1.0f / (1.0f + __expf(-val));
                Out[(mb + v) * N + nb] = val;
            }
        }
}

// ---------------------------------------------------------------------------
// Chunked parallel scan: s_t = lam_t * s_{t-1} + u_t per (b, r) chain.
// ---------------------------------------------------------------------------
__global__ __launch_bounds__(256) void scan_chunk(
    const float* __restrict__ lam, const float* __restrict__ u,
    float* __restrict__ cA, float* __restrict__ cS)
{
    int g = blockIdx.x * 256 + threadIdx.x;  // g = b*4096 + c*64 + r
    int r = g & 63, c = (g >> 6) & 63, b = g >> 12;
    int base = (b * Tv + c * CHKL) * PRv + r;
    float Aacc = 1.0f, s = 0.0f;
    for (int i = 0; i < CHKL; ++i) {
        float l = lam[base + i * PRv];
        float uu = u[base + i * PRv];
        s = l * s + uu;
        Aacc *= l;
    }
    cA[g] = Aacc;
    cS[g] = s;
}

__global__ __launch_bounds__(256) void scan_carry(
    const float* __restrict__ cA, const float* __restrict__ cS,
    float* __restrict__ carry)
{
    int g = threadIdx.x;       // 256 = B*PR chains
    int r = g & 63, b = g >> 6;
    float s = 0.0f;
    for (int c = 0; c < NCHK; ++c) {
        int idx = (b * NCHK + c) * PRv + r;
        carry[idx] = s;
        s = cA[idx] * s + cS[idx];
    }
}

__global__ __launch_bounds__(256) void scan_apply(
    const float* __restrict__ lam, const float* __restrict__ u,
    const float* __restrict__ carry, float* __restrict__ sseq)
{
    int g = blockIdx.x * 256 + threadIdx.x;
    int r = g & 63, c = (g >> 6) & 63, b = g >> 12;
    int base = (b * Tv + c * CHKL) * PRv + r;
    float s = carry[g];
    for (int i = 0; i < CHKL; ++i) {
        int idx = base + i * PRv;
        s = lam[idx] * s + u[idx];
        sseq[idx] = s;
    }
}

// ---------------------------------------------------------------------------
extern "C" void kernel_launch(void* const* d_in, const int* in_sizes, int n_in,
                              void* d_out, int out_size, void* d_ws, size_t ws_size,
                              hipStream_t stream)
{
    const float* x      = (const float*)d_in[0];
    const float* WtoP   = (const float*)d_in[1];
    const float* WU     = (const float*)d_in[2];
    const float* WF     = (const float*)d_in[3];
    const float* WV     = (const float*)d_in[4];
    const float* Wlam   = (const float*)d_in[5];
    const float* Bmat   = (const float*)d_in[6];
    const float* Cmat   = (const float*)d_in[7];
    const float* WfromP = (const float*)d_in[8];
    float* out = (float*)d_out;

    float* We  = (float*)d_ws;                        // D*P
    float* t   = We  + (size_t)Dv * Pv;               // M*P (becomes t_tilde in place)
    float* Fg  = t   + (size_t)Mv * Pv;               // M*P
    float* lam = Fg  + (size_t)Mv * Pv;               // M*PR
    float* u   = lam + (size_t)Mv * PRv;              // M*PR
    float* ss  = u   + (size_t)Mv * PRv;              // M*PR
    float* cA  = ss  + (size_t)Mv * PRv;              // B*NCHK*PR
    float* cS  = cA  + (size_t)Bv * NCHK * PRv;
    float* cy  = cS  + (size_t)Bv * NCHK * PRv;

    const dim3 blk(256);

    // 0) fold W_U * W_V into W_eff
    weff_kernel<<<(Dv * Pv) / 256, blk, 0, stream>>>(WtoP, WU, WV, We);

    // 1) t = x @ W_eff          (M=16384, N=512, K=1024)
    gemm_wmma<ACT_NONE, false><<<dim3(Pv / BN, Mv / BM), blk, 0, stream>>>(
        x, We, nullptr, t, Mv, Pv, Dv);
    // 2) Fg = sigmoid(x @ W_F)
    gemm_wmma<ACT_SIGMOID, false><<<dim3(Pv / BN, Mv / BM), blk, 0, stream>>>(
        x, WF, nullptr, Fg, Mv, Pv, Dv);
    // 3) lam = sigmoid(Fg @ W_lam)   (N=64, K=512)
    gemm_wmma<ACT_SIGMOID, false><<<dim3(PRv / BN, Mv / BM), blk, 0, stream>>>(
        Fg, Wlam, nullptr, lam, Mv, PRv, Pv);
    // 4) u = t @ B_mat               (N=64, K=512)
    gemm_wmma<ACT_NONE, false><<<dim3(PRv / BN, Mv / BM), blk, 0, stream>>>(
        t, Bmat, nullptr, u, Mv, PRv, Pv);

    // 5) chunked scan over T
    scan_chunk<<<(Bv * NCHK * PRv) / 256, blk, 0, stream>>>(lam, u, cA, cS);
    scan_carry<<<1, blk, 0, stream>>>(cA, cS, cy);
    scan_apply<<<(Bv * NCHK * PRv) / 256, blk, 0, stream>>>(lam, u, cy, ss);

    // 6) t_tilde = t + ss @ C_mat    (N=512, K=64, accumulate into t in place)
    gemm_wmma<ACT_NONE, true><<<dim3(Pv / BN, Mv / BM), blk, 0, stream>>>(
        ss, Cmat, t, t, Mv, Pv, PRv);
    // 7) h = t_tilde @ W_fromP       (N=1024, K=512) -> d_out
    gemm_wmma<ACT_NONE, false><<<dim3(Dv / BN, Mv / BM), blk, 0, stream>>>(
        t, WfromP, nullptr, out, Mv, Dv, Pv);
}